// SGC_23553600651662
// MI455X (gfx1250) — compile-verified
//
#include <hip/hip_runtime.h>
#include <math.h>

typedef float v2f __attribute__((ext_vector_type(2)));
typedef float v8f __attribute__((ext_vector_type(8)));

#define DIN  128
#define DOUT 64

// ---------------------------------------------------------------------------
// Hardware fp32 atomic add (relaxed, device scope) -> global_atomic_add_f32
// ---------------------------------------------------------------------------
__device__ __forceinline__ void atomic_add_f32(float* p, float v) {
  __hip_atomic_fetch_add(p, v, __ATOMIC_RELAXED, __HIP_MEMORY_SCOPE_AGENT);
}

// ---------------------------------------------------------------------------
// gcn_norm stage
// ---------------------------------------------------------------------------
__global__ __launch_bounds__(256) void k_init_deg(float* __restrict__ deg, int N) {
  int i = blockIdx.x * blockDim.x + threadIdx.x;
  if (i < N) deg[i] = 1.0f;  // self-loop weight (fill_value = 1.0)
}

__global__ __launch_bounds__(256) void k_accum_deg(const int* __restrict__ dst,
                                                   const float* __restrict__ w,
                                                   float* __restrict__ deg, int E) {
  int e = blockIdx.x * blockDim.x + threadIdx.x;
  if (e < E) atomic_add_f32(&deg[dst[e]], w[e]);
}

__global__ __launch_bounds__(256) void k_dinv(float* __restrict__ deg, int N) {
  int i = blockIdx.x * blockDim.x + threadIdx.x;
  if (i < N) {
    float d = deg[i];
    deg[i] = (d > 0.0f) ? rsqrtf(d) : 0.0f;   // deg >= 1 always (self-loop)
  }
}

__global__ __launch_bounds__(256) void k_norm(const int* __restrict__ src,
                                              const int* __restrict__ dst,
                                              const float* __restrict__ w,
                                              const float* __restrict__ dinv,
                                              float* __restrict__ nrm, int E) {
  int e = blockIdx.x * blockDim.x + threadIdx.x;
  if (e < E) nrm[e] = dinv[src[e]] * w[e] * dinv[dst[e]];
}

// ---------------------------------------------------------------------------
// Propagation hop: h_out = dinv^2 * h_in (self-loop) + scatter(norm * h_in[src])
// ---------------------------------------------------------------------------
__global__ __launch_bounds__(256) void k_prop_init(const float4* __restrict__ hin,
                                                   const float* __restrict__ dinv,
                                                   float4* __restrict__ hout, int total) {
  int t = blockIdx.x * blockDim.x + threadIdx.x;
  if (t < total) {
    int i = t >> 5;              // 32 float4 per node row (DIN=128)
    float s = dinv[i];
    s = s * s;                   // self-loop norm = dinv[i]*1.0*dinv[i]
    float4 v = hin[t];
    hout[t] = make_float4(s * v.x, s * v.y, s * v.z, s * v.w);
  }
}

// One wave (32 lanes) per edge; each lane handles a float4 chunk of the row.
// h (51 MB) is L2-resident (192 MB L2) so gathers + atomics stay on-chip.
__global__ __launch_bounds__(256) void k_prop_scatter(const int* __restrict__ src,
                                                      const int* __restrict__ dst,
                                                      const float* __restrict__ nrm,
                                                      const float* __restrict__ hin,
                                                      float* __restrict__ hout, int E) {
  const int lane = threadIdx.x & 31;
  const int wib  = threadIdx.x >> 5;
  const int estride = gridDim.x * 8;          // 8 waves per 256-thread block
  for (int e = blockIdx.x * 8 + wib; e < E; e += estride) {
    int   s  = src[e];
    int   d  = dst[e];
    float nm = nrm[e];
    // Prefetch next source row into cache (global_prefetch_b8)
    if (e + estride < E) {
      int sn = src[e + estride];
      __builtin_prefetch(hin + (size_t)sn * DIN + lane * 4, 0, 0);
    }
    const float4 hv = *(const float4*)(hin + (size_t)s * DIN + lane * 4);
    float* o = hout + (size_t)d * DIN + lane * 4;
    atomic_add_f32(o + 0, nm * hv.x);
    atomic_add_f32(o + 1, nm * hv.y);
    atomic_add_f32(o + 2, nm * hv.z);
    atomic_add_f32(o + 3, nm * hv.w);
  }
}

// ---------------------------------------------------------------------------
// Fused GEMM (fp32 WMMA 16x16x4) + bias + ReLU + row-wise log_softmax.
// One wave computes a 16x64 output tile (4 col-tiles x 32 k-steps).
// ---------------------------------------------------------------------------
#define LDS_STRIDE 68  // 16-col groups offset by 8 rows land on disjoint banks

__global__ __launch_bounds__(256) void k_gemm_lsm(const float* __restrict__ h,
                                                  const float* __restrict__ Wm,
                                                  const float* __restrict__ bias,
                                                  float* __restrict__ out, int N) {
  __shared__ float lds[8][16 * LDS_STRIDE];

  const int lane = threadIdx.x & 31;
  const int wave = threadIdx.x >> 5;
  const int half = lane >> 4;        // 0: lanes 0-15, 1: lanes 16-31
  const int m16  = lane & 15;
  const int kadj = half * 2;         // A/B fragment K offset for upper half-wave

  const long long row0 = ((long long)blockIdx.x * 8 + wave) * 16;
  long long arow = row0 + m16;
  if (arow >= N) arow = (long long)N - 1;     // clamp; stores predicated later
  const float* __restrict__ ap = h + arow * DIN;

  v8f acc0 = {}, acc1 = {}, acc2 = {}, acc3 = {};

  #pragma unroll
  for (int k0 = 0; k0 < DIN; k0 += 4) {
    // A fragment 16x4 f32: lane m holds A[m][k0+kadj], A[m][k0+kadj+1]
    v2f a;
    a.x = ap[k0 + kadj + 0];
    a.y = ap[k0 + kadj + 1];
    // B fragments 4x16 f32 per column tile (mirror layout)
    const float* wp0 = Wm + (size_t)(k0 + kadj + 0) * DOUT + m16;
    const float* wp1 = Wm + (size_t)(k0 + kadj + 1) * DOUT + m16;
    v2f b0; b0.x = wp0[0];  b0.y = wp1[0];
    v2f b1; b1.x = wp0[16]; b1.y = wp1[16];
    v2f b2; b2.x = wp0[32]; b2.y = wp1[32];
    v2f b3; b3.x = wp0[48]; b3.y = wp1[48];
    acc0 = __builtin_amdgcn_wmma_f32_16x16x4_f32(false, a, false, b0, (short)0, acc0, false, false);
    acc1 = __builtin_amdgcn_wmma_f32_16x16x4_f32(false, a, false, b1, (short)0, acc1, false, false);
    acc2 = __builtin_amdgcn_wmma_f32_16x16x4_f32(false, a, false, b2, (short)0, acc2, false, false);
    acc3 = __builtin_amdgcn_wmma_f32_16x16x4_f32(false, a, false, b3, (short)0, acc3, false, false);
  }

  // Epilogue: bias + ReLU into wave-private LDS tile (row = r + 8*half)
  v8f accs[4] = {acc0, acc1, acc2, acc3};
  float* my = lds[wave];
  #pragma unroll
  for (int ct = 0; ct < 4; ++ct) {
    float bb = bias[ct * 16 + m16];
    #pragma unroll
    for (int r = 0; r < 8; ++r) {
      float v = accs[ct][r] + bb;
      my[(r + half * 8) * LDS_STRIDE + ct * 16 + m16] = (v > 0.0f) ? v : 0.0f;
    }
  }

  // Row-wise log_softmax over 64 cols: 2 values/lane + wave shuffle reductions
  for (int r = 0; r < 16; ++r) {
    float v0 = my[r * LDS_STRIDE + lane];
    float v1 = my[r * LDS_STRIDE + 32 + lane];
    float mx = fmaxf(v0, v1);
    #pragma unroll
    for (int o = 16; o >= 1; o >>= 1) mx = fmaxf(mx, __shfl_xor(mx, o, 32));
    float se = __expf(v0 - mx) + __expf(v1 - mx);
    #pragma unroll
    for (int o = 16; o >= 1; o >>= 1) se += __shfl_xor(se, o, 32);
    float lse = mx + __logf(se);
    long long gr = row0 + r;
    if (gr < N) {
      out[gr * DOUT + lane]      = v0 - lse;
      out[gr * DOUT + 32 + lane] = v1 - lse;
    }
  }
}

// ---------------------------------------------------------------------------
// Launch
// ---------------------------------------------------------------------------
extern "C" void kernel_launch(void* const* d_in, const int* in_sizes, int n_in,
                              void* d_out, int out_size, void* d_ws, size_t ws_size,
                              hipStream_t stream) {
  const float* x  = (const float*)d_in[0];
  const int*   ei = (const int*)d_in[1];     // edge_index, shape (2, E)
  const float* ew = (const float*)d_in[2];
  const float* Wm = (const float*)d_in[3];
  const float* bv = (const float*)d_in[4];

  const int N = in_sizes[0] / DIN;
  const int E = in_sizes[2];
  const int* src = ei;
  const int* dst = ei + E;

  // Carve workspace: dinv | norm | h1 | h2
  char*  ws  = (char*)d_ws;
  size_t off = 0;
  auto carve = [&](size_t bytes) -> void* {
    void* p = ws + off;
    off = (off + bytes + 255) & ~(size_t)255;
    return p;
  };
  float* dinv = (float*)carve((size_t)N * sizeof(float));
  float* nrm  = (float*)carve((size_t)E * sizeof(float));
  float* h1   = (float*)carve((size_t)N * DIN * sizeof(float));
  float* h2   = (float*)carve((size_t)N * DIN * sizeof(float));
  (void)ws_size;

  const int T = 256;
  // gcn_norm
  k_init_deg <<<(N + T - 1) / T, T, 0, stream>>>(dinv, N);
  k_accum_deg<<<(E + T - 1) / T, T, 0, stream>>>(dst, ew, dinv, E);
  k_dinv     <<<(N + T - 1) / T, T, 0, stream>>>(dinv, N);
  k_norm     <<<(E + T - 1) / T, T, 0, stream>>>(src, dst, ew, dinv, nrm, E);

  const int NC = N * 32;   // float4 chunks
  // hop 1: x -> h1
  k_prop_init   <<<(NC + T - 1) / T, T, 0, stream>>>((const float4*)x, dinv, (float4*)h1, NC);
  k_prop_scatter<<<2048, T, 0, stream>>>(src, dst, nrm, x, h1, E);
  // hop 2: h1 -> h2
  k_prop_init   <<<(NC + T - 1) / T, T, 0, stream>>>((const float4*)h1, dinv, (float4*)h2, NC);
  k_prop_scatter<<<2048, T, 0, stream>>>(src, dst, nrm, h1, h2, E);

  // fused GEMM + bias + ReLU + log_softmax
  k_gemm_lsm<<<(N + 127) / 128, T, 0, stream>>>(h2, Wm, bv, (float*)d_out, N);
}